// edgeConv_70428873719839
// MI455X (gfx1250) — compile-verified
//
#include <hip/hip_runtime.h>
#include <hip/hip_bf16.h>

typedef float v2f __attribute__((ext_vector_type(2)));
typedef float v8f __attribute__((ext_vector_type(8)));

#define BATCH 4
#define NPTS  4096
#define FDIM  64
#define KNN   20
#define CJ    64   // candidate chunk per LDS stage
#define BQ    64   // query rows per workgroup (kernel 2)

// ---------------------------------------------------------------------------
// Kernel 1: yz[row][0:64]  = y = W1 . x   (W[:, :64])
//           yz[row][64:128]= z = W2 . x   (W[:, 64:])
//           xs[row] = ||x_row||^2
// One wave per 16-row M-tile, WMMA f32 16x16x4, K=64 in 16 steps.
// ---------------------------------------------------------------------------
__global__ __launch_bounds__(128)
void k_gemm_yz(const float* __restrict__ x, const float* __restrict__ W,
               float* __restrict__ yz, float* __restrict__ xs)
{
    const int lane = threadIdx.x & 31;
    const int wave = threadIdx.x >> 5;
    const int half = lane >> 4;      // 0: K{0,1}, 1: K{2,3}
    const int l16  = lane & 15;
    const int row0 = blockIdx.x * 64 + wave * 16;   // global row (b*N + n)

    // A fragments: 16x4 f32 tiles, one float2 per lane per k-step
    v2f a[16];
    const float* xrow = x + (size_t)(row0 + l16) * FDIM + 2 * half;
#pragma unroll
    for (int s = 0; s < 16; ++s) a[s] = *(const v2f*)(xrow + 4 * s);

    // squared norm: lane L and L+16 hold complementary K halves of row l16
    float ss = 0.f;
#pragma unroll
    for (int s = 0; s < 16; ++s) ss += a[s].x * a[s].x + a[s].y * a[s].y;
    ss += __shfl_xor(ss, 16, 32);
    if (lane < 16) xs[row0 + l16] = ss;

    // 8 N-tiles of 16 output columns (128 total: y then z)
#pragma unroll
    for (int nt = 0; nt < 8; ++nt) {
        const int o = nt * 16 + l16;                 // 0..127
        // B[k][o]: o<64 -> W[o][k] ; o>=64 -> W[o-64][64+k]
        const float* wrow = (o < 64) ? (W + (size_t)o * 128 + 2 * half)
                                     : (W + (size_t)(o - 64) * 128 + 64 + 2 * half);
        v8f c = {};
#pragma unroll
        for (int s = 0; s < 16; ++s) {
            v2f bf = *(const v2f*)(wrow + 4 * s);
            c = __builtin_amdgcn_wmma_f32_16x16x4_f32(
                    false, a[s], false, bf, (short)0, c, false, false);
        }
#pragma unroll
        for (int r = 0; r < 8; ++r) {
            const int m = r + 8 * half;
            yz[(size_t)(row0 + m) * 128 + o] = c[r];
        }
    }
}

// ---------------------------------------------------------------------------
// Kernel 2: per 64-query tile of one batch:
//   - double-buffered GLOBAL_LOAD_ASYNC_TO_LDS_B128 candidate streaming
//   - WMMA f32 16x16x4 distance tiles vs all 4096 candidates
//   - per-thread top-20 selection on order-value (xs_j - 2*dot)
//   - epilogue: out = y + bias - z_self + max_{j in top20} z_j
// Block = 64 threads = 2 waves; wave w owns query row-tiles {2w, 2w+1}.
//
// Dynamic LDS layout (floats):
//   buf0  : [    0,  4096)   candidate chunk A   (16 KB)
//   buf1  : [ 4096,  8192)   candidate chunk B   (16 KB)
//   sdist : [ 8192, 12352)   [j_local][m] padded 64x65
//   sxs   : [12352, 12480)   candidate sq-norms, double buffered 2x64
// ---------------------------------------------------------------------------
#define LDS_BUF0  0
#define LDS_BUF1  4096
#define LDS_DIST  8192
#define LDS_SXS   12352
#define LDS_TOTAL_BYTES ((12352 + 128) * 4)

__global__ __launch_bounds__(64)
void k_knn_out(const float* __restrict__ x, const float* __restrict__ bias,
               const float* __restrict__ yz, const float* __restrict__ xs,
               float* __restrict__ out)
{
    extern __shared__ float smem[];

    const int tid  = threadIdx.x;
    const int lane = tid & 31;
    const int wave = tid >> 5;
    const int half = lane >> 4;
    const int l16  = lane & 15;
    const int b     = blockIdx.x >> 6;      // N/BQ = 64 tiles per batch
    const int n0    = (blockIdx.x & 63) * BQ;
    const size_t rb = (size_t)b * NPTS;

    // A fragments for this wave's two 16-row query tiles (kept in VGPRs)
    v2f afrag[2][16];
#pragma unroll
    for (int rt = 0; rt < 2; ++rt) {
        const float* xr = x + (rb + n0 + (wave * 2 + rt) * 16 + l16) * FDIM + 2 * half;
#pragma unroll
        for (int s = 0; s < 16; ++s) afrag[rt][s] = *(const v2f*)(xr + 4 * s);
    }

    float dbest[KNN];
    int   jbest[KNN];
#pragma unroll
    for (int i = 0; i < KNN; ++i) { dbest[i] = 3.4e38f; jbest[i] = 0; }

    // issue one candidate chunk (CJ rows x FDIM floats) as async B128 copies
    auto issue_chunk = [&](int j0, int buf) {
        const unsigned ldsbase = (unsigned)((buf ? LDS_BUF1 : LDS_BUF0) * 4);
#pragma unroll
        for (int it = 0; it < (CJ * FDIM / 4) / BQ; ++it) {   // 16 iters
            const int i  = it * BQ + tid;
            const int jl = i >> 4;
            const int kk = (i & 15) * 4;
            const unsigned lds_off = ldsbase + (unsigned)(jl * FDIM + kk) * 4u;
            const float* ga = x + (rb + j0 + jl) * FDIM + kk;
            asm volatile("global_load_async_to_lds_b128 %0, %1, off"
                         :: "v"(lds_off), "v"(ga) : "memory");
        }
        // candidate squared norms (tiny, regular path)
        smem[LDS_SXS + buf * CJ + tid] = xs[rb + j0 + tid];
    };

    issue_chunk(0, 0);

    for (int c = 0; c < NPTS / CJ; ++c) {
        const int j0  = c * CJ;
        const int cur = c & 1;

        // drain this wave's async transfers, then make chunk visible WG-wide
        asm volatile("s_wait_asynccnt 0x0" ::: "memory");
        __syncthreads();

        // start streaming the next chunk into the other buffer
        if (j0 + CJ < NPTS) issue_chunk(j0 + CJ, 1 - cur);

        const float* bufp = smem + (cur ? LDS_BUF1 : LDS_BUF0);
        const float* sxsp = smem + LDS_SXS + cur * CJ;

        // WMMA over 4 candidate column-tiles of 16
        for (int ct = 0; ct < CJ / 16; ++ct) {
            v2f bfrag[16];
            const float* bj = bufp + (ct * 16 + l16) * FDIM + 2 * half;
#pragma unroll
            for (int s = 0; s < 16; ++s) bfrag[s] = *(const v2f*)(bj + 4 * s);
            const float xsj = sxsp[ct * 16 + l16];
#pragma unroll
            for (int rt = 0; rt < 2; ++rt) {
                v8f cacc = {};
#pragma unroll
                for (int s = 0; s < 16; ++s)
                    cacc = __builtin_amdgcn_wmma_f32_16x16x4_f32(
                            false, afrag[rt][s], false, bfrag[s],
                            (short)0, cacc, false, false);
#pragma unroll
                for (int r = 0; r < 8; ++r) {
                    const int m = (wave * 2 + rt) * 16 + r + 8 * half;
                    smem[LDS_DIST + (ct * 16 + l16) * (BQ + 1) + m] =
                        xsj - 2.0f * cacc[r];
                }
            }
        }
        __syncthreads();

        // selection: thread t owns query row n0+t
        const int nself = n0 + tid;
        for (int jl = 0; jl < CJ; ++jl) {
            const int j = j0 + jl;
            if (j == nself) continue;
            const float d = smem[LDS_DIST + jl * (BQ + 1) + tid];
            if (d < dbest[KNN - 1]) {
                int p = KNN - 1;
                while (p > 0 && dbest[p - 1] > d) {
                    dbest[p] = dbest[p - 1]; jbest[p] = jbest[p - 1]; --p;
                }
                dbest[p] = d; jbest[p] = j;
            }
        }
        __syncthreads();   // sdist free for next chunk; buffers swap
    }

    // epilogue: out = y + bias - z_self + max_k z_j
    const float* yzrow = yz + (rb + n0 + tid) * 128;
    float*       orow  = out + (rb + n0 + tid) * FDIM;
    for (int c0 = 0; c0 < FDIM; c0 += 4) {
        float4 acc = make_float4(-3.4e38f, -3.4e38f, -3.4e38f, -3.4e38f);
#pragma unroll
        for (int i = 0; i < KNN; ++i) {
            const float4 zj = *(const float4*)(yz + (rb + jbest[i]) * 128 + 64 + c0);
            acc.x = fmaxf(acc.x, zj.x); acc.y = fmaxf(acc.y, zj.y);
            acc.z = fmaxf(acc.z, zj.z); acc.w = fmaxf(acc.w, zj.w);
        }
        const float4 y4 = *(const float4*)(yzrow + c0);
        const float4 z4 = *(const float4*)(yzrow + 64 + c0);
        const float4 b4 = *(const float4*)(bias + c0);
        float4 o4;
        o4.x = y4.x + b4.x - z4.x + acc.x;
        o4.y = y4.y + b4.y - z4.y + acc.y;
        o4.z = y4.z + b4.z - z4.z + acc.z;
        o4.w = y4.w + b4.w - z4.w + acc.w;
        *(float4*)(orow + c0) = o4;
    }
}

// ---------------------------------------------------------------------------
extern "C" void kernel_launch(void* const* d_in, const int* in_sizes, int n_in,
                              void* d_out, int out_size, void* d_ws, size_t ws_size,
                              hipStream_t stream)
{
    const float* x    = (const float*)d_in[0];   // [4,4096,64]
    const float* W    = (const float*)d_in[1];   // [64,128]
    const float* bias = (const float*)d_in[2];   // [64]
    (void)in_sizes; (void)n_in; (void)out_size; (void)ws_size;

    float* yz = (float*)d_ws;                          // [B*N][128] : 8 MB
    float* xs = yz + (size_t)BATCH * NPTS * 128;       // [B*N]      : 64 KB
    float* out = (float*)d_out;                        // [B,N,64]

    // K1: point-wise GEMM (y|z) + squared norms
    k_gemm_yz<<<(BATCH * NPTS) / 64, 128, 0, stream>>>(x, W, yz, xs);
    // K2: async-staged WMMA kNN distances + top-20 + gather-max epilogue
    k_knn_out<<<BATCH * (NPTS / BQ), 64, LDS_TOTAL_BYTES, stream>>>(
        x, bias, yz, xs, out);
}